// HPGCN_17119739641940
// MI455X (gfx1250) — compile-verified
//
#include <hip/hip_runtime.h>

typedef __attribute__((ext_vector_type(2))) float v2f;
typedef __attribute__((ext_vector_type(8))) float v8f;

#define N_NODES  50000
#define N_FEAT   128
#define N_HIDDEN 256
#define N_EDGES  800000
#define N_PAD    50176   // N_NODES rounded up to 256

// --- degree / normalization -------------------------------------------------

__global__ void k_init_deg(float* __restrict__ deg) {
  int i = blockIdx.x * blockDim.x + threadIdx.x;
  if (i < N_NODES) deg[i] = 1.0f;            // self loop
}

__global__ void k_count_deg(const int* __restrict__ dst, float* __restrict__ deg) {
  int e = blockIdx.x * blockDim.x + threadIdx.x;
  if (e < N_EDGES) atomicAdd(&deg[dst[e]], 1.0f);
}

__global__ void k_inv_sqrt(float* __restrict__ inv) {
  int i = blockIdx.x * blockDim.x + threadIdx.x;
  if (i < N_NODES) inv[i] = rsqrtf(inv[i]);  // deg >= 1 always
}

// --- layer-1 aggregation over raw features (A-hat @ X), 128 wide ------------

// self-loop contribution: agg1[i,f] = x[i,f] * inv[i]^2
__global__ void k_agg1_init(const float* __restrict__ x, const float* __restrict__ inv,
                            float* __restrict__ agg1) {
  int idx = blockIdx.x * blockDim.x + threadIdx.x;   // over N_NODES*N_FEAT
  int i = idx >> 7;                                  // /128
  float w = inv[i];
  agg1[idx] = x[idx] * (w * w);
}

// one wave per edge; lane handles 4 consecutive features (float4 gather + f32 atomics)
__global__ void k_agg1_edges(const float* __restrict__ x, const int* __restrict__ src,
                             const int* __restrict__ dst, const float* __restrict__ inv,
                             float* __restrict__ agg1) {
  int wave = (int)((blockIdx.x * blockDim.x + threadIdx.x) >> 5);
  int lane = threadIdx.x & 31;
  if (wave >= N_EDGES) return;
  int s = src[wave], d = dst[wave];
  float w = inv[s] * inv[d];
  const float4* xs = (const float4*)(x + (size_t)s * N_FEAT);
  float4 v = xs[lane];
  float* o = agg1 + (size_t)d * N_FEAT + lane * 4;
  atomicAdd(o + 0, v.x * w);
  atomicAdd(o + 1, v.y * w);
  atomicAdd(o + 2, v.z * w);
  atomicAdd(o + 3, v.w * w);
}

// --- layer-1 GEMM: H = relu(agg1 @ W1 + b1) via V_WMMA_F32_16X16X4_F32 ------

// block = 512 threads = 16 waves; wave w computes tile (blockIdx.x, w) of [3125 x 16] tiles
__global__ void __launch_bounds__(512)
k_gemm1_wmma(const float* __restrict__ A,   // [N_NODES, 128]  (agg1)
             const float* __restrict__ W,   // [128, 256]
             const float* __restrict__ b,   // [256]
             float* __restrict__ H) {       // [N_NODES, 256]
  int wave = threadIdx.x >> 5;      // 0..15 -> n tile
  int lane = threadIdx.x & 31;
  int m0 = blockIdx.x * 16;
  int n0 = wave * 16;
  int half = lane >> 4;             // 0: K={k,k+1}, 1: K={k+2,k+3}
  int l    = lane & 15;
  int row  = m0 + l;                // A-fragment row (M = lane%16)
  int col  = n0 + l;                // B/C/D column (N = lane%16)

  const float* arow = A + (size_t)row * N_FEAT;
  v8f c = {};
  for (int k = 0; k < N_FEAT; k += 4) {
    int ka = k + 2 * half;
    v2f a, bb;
    a.x  = arow[ka];
    a.y  = arow[ka + 1];
    bb.x = W[(size_t)ka * N_HIDDEN + col];
    bb.y = W[(size_t)(ka + 1) * N_HIDDEN + col];
    // D = A(16x4,f32) * B(4x16,f32) + C ; 8-arg form:
    // (neg_a, A, neg_b, B, c_mod, C, reuse_a, reuse_b)
    c = __builtin_amdgcn_wmma_f32_16x16x4_f32(
        false, a, false, bb, (short)0, c, false, false);
  }

  float bias = b[col];
  for (int r = 0; r < 8; ++r) {
    int orow = m0 + r + 8 * half;   // C/D layout: VGPR r, lanes 16-31 hold M+8
    float v = c[r] + bias;
    H[(size_t)orow * N_HIDDEN + col] = v > 0.0f ? v : 0.0f;
  }
}

// --- layer-2: z = H @ W2 ([N,1]); wave per node, shuffle reduce -------------

__global__ void k_dot_w2(const float* __restrict__ H, const float* __restrict__ W2,
                         float* __restrict__ z) {
  int wave = (int)((blockIdx.x * blockDim.x + threadIdx.x) >> 5);
  int lane = threadIdx.x & 31;
  if (wave >= N_NODES) return;
  const float* h = H + (size_t)wave * N_HIDDEN;
  float acc = 0.0f;
#pragma unroll
  for (int j = 0; j < N_HIDDEN / 32; ++j)
    acc += h[lane + 32 * j] * W2[lane + 32 * j];
  for (int off = 16; off > 0; off >>= 1)
    acc += __shfl_xor(acc, off, 32);
  if (lane == 0) z[wave] = acc;
}

// --- layer-2 scalar aggregation ---------------------------------------------

__global__ void k_out_init(const float* __restrict__ z, const float* __restrict__ inv,
                           const float* __restrict__ b2, float* __restrict__ out) {
  int i = blockIdx.x * blockDim.x + threadIdx.x;
  if (i < N_NODES) {
    float w = inv[i];
    out[i] = z[i] * (w * w) + b2[0];
  }
}

__global__ void k_out_edges(const int* __restrict__ src, const int* __restrict__ dst,
                            const float* __restrict__ inv, const float* __restrict__ z,
                            float* __restrict__ out) {
  int e = blockIdx.x * blockDim.x + threadIdx.x;
  if (e < N_EDGES) {
    int s = src[e], d = dst[e];
    atomicAdd(&out[d], z[s] * inv[s] * inv[d]);
  }
}

// --- launch ------------------------------------------------------------------

extern "C" void kernel_launch(void* const* d_in, const int* in_sizes, int n_in,
                              void* d_out, int out_size, void* d_ws, size_t ws_size,
                              hipStream_t stream) {
  const float* x   = (const float*)d_in[0];
  const int*   ei  = (const int*)d_in[1];
  const float* W1  = (const float*)d_in[2];
  const float* b1  = (const float*)d_in[3];
  const float* W2  = (const float*)d_in[4];
  const float* b2  = (const float*)d_in[5];
  const int* src = ei;
  const int* dst = ei + N_EDGES;
  float* out = (float*)d_out;

  float* ws   = (float*)d_ws;
  float* inv  = ws;                          // [N_PAD]      (deg, then inv_sqrt in place)
  float* z    = ws + N_PAD;                  // [N_PAD]
  float* agg1 = ws + 2 * (size_t)N_PAD;      // [N_NODES*128]
  float* h1   = agg1 + (size_t)N_NODES * N_FEAT; // [N_NODES*256]

  const int T = 256;

  // 1) degrees + inverse sqrt
  k_init_deg  <<<(N_NODES + T - 1) / T, T, 0, stream>>>(inv);
  k_count_deg <<<(N_EDGES + T - 1) / T, T, 0, stream>>>(dst, inv);
  k_inv_sqrt  <<<(N_NODES + T - 1) / T, T, 0, stream>>>(inv);

  // 2) agg1 = A-hat @ X  (self-loop init + per-edge scatter, 1 wave/edge)
  k_agg1_init <<<(N_NODES * N_FEAT) / T, T, 0, stream>>>(x, inv, agg1);
  k_agg1_edges<<<(N_EDGES * 32) / T, T, 0, stream>>>(x, src, dst, inv, agg1);

  // 3) H = relu(agg1 @ W1 + b1)  -- WMMA f32 16x16x4, 3125 M-tiles x 16 N-tiles
  k_gemm1_wmma<<<N_NODES / 16, 512, 0, stream>>>(agg1, W1, b1, h1);

  // 4) z = H @ W2  (wave per node)
  k_dot_w2    <<<(N_NODES * 32 + T - 1) / T, T, 0, stream>>>(h1, W2, z);

  // 5) out = A-hat @ z + b2
  k_out_init  <<<(N_NODES + T - 1) / T, T, 0, stream>>>(z, inv, b2, out);
  k_out_edges <<<(N_EDGES + T - 1) / T, T, 0, stream>>>(src, dst, inv, z, out);
}